// myGATConv_53309134078256
// MI455X (gfx1250) — compile-verified
//
#include <hip/hip_runtime.h>

// ---------------- problem constants (match reference) ----------------
#define N_NODES   100000
#define N_EDGES   1600000
#define IN_DIM    256
#define H_HEADS   4
#define F_DIM     32
#define HF        128        // H*F
#define EF_DIM    64
#define NTYPES    8
#define NEG_SLOPE 0.2f

typedef float v2f __attribute__((ext_vector_type(2)));
typedef float v8f __attribute__((ext_vector_type(8)));
typedef unsigned int u32x4 __attribute__((ext_vector_type(4)));
typedef int i32x4 __attribute__((ext_vector_type(4)));
typedef int i32x8 __attribute__((ext_vector_type(8)));

#if defined(__gfx1250__) && __has_builtin(__builtin_amdgcn_wmma_f32_16x16x4_f32)
#define USE_WMMA 1
#else
#define USE_WMMA 0
#endif

#if defined(__gfx1250__) && __has_builtin(__builtin_amdgcn_tensor_load_to_lds) && \
    __has_builtin(__builtin_amdgcn_s_wait_tensorcnt)
#define USE_TDM 1
#else
#define USE_TDM 0
#endif

// ---------------- GEMM tiling ----------------
constexpr int KC = 16;     // K chunk staged in LDS
constexpr int PA = 20;     // As pitch = KC + 4 pad  (20*d % 64 == 0 mod 4 -> no A-frag conflicts)
constexpr int PB = 136;    // Bs pitch = HF + 8 pad  (2*136 % 64 == 16 -> halves 16 banks apart)
constexpr int NCHUNK = IN_DIM / KC;                 // 16
constexpr int BUF_FLOATS = 128 * PA + KC * PB;      // 2560 + 2176 = 4736 floats = 18944 B
// dynamic LDS = 2 * BUF_FLOATS * 4 = 37888 B (double buffered)

// ---------------------------------------------------------------------
// TDM: 2D tile load Global -> LDS via Tensor Data Mover descriptor.
//   D# group0: [1:0]=count=1, [63:32]=lds_addr, [120:64]=global_addr, [127:126]=type(2)
//   D# group1: [17:16]=data_size(2=4B), [20]=pad_enable, [24:22]=pad_interval code
//              (2^(c+1) DWORDs), [31:25]=pad_amount code (c+1 DWORDs),
//              [79:48]=tensor_dim0, [111:80]=tensor_dim1, [127:112]=tile_dim0,
//              [143:128]=tile_dim1, [207:160]=tensor_dim0_stride
// OOB rows (tensor_dim1) read as zero -> handles ragged last row-block.
// ---------------------------------------------------------------------
#if USE_TDM
__device__ __forceinline__ void tdm_load_2d(unsigned lds_off, const void* gptr,
                                            unsigned tensor_d0, unsigned tensor_d1,
                                            unsigned tile_d0, unsigned tile_d1,
                                            unsigned row_stride,
                                            unsigned pad_interval_code,
                                            unsigned pad_amount_code) {
  unsigned long long ga = (unsigned long long)(uintptr_t)gptr;
  u32x4 g0;
  g0.x = 1u;                                        // count=1 (valid user descriptor)
  g0.y = lds_off;                                   // LDS byte address
  g0.z = (unsigned)(ga & 0xFFFFFFFFu);              // global_addr[31:0]
  g0.w = (unsigned)((ga >> 32) & 0x01FFFFFFu) | (2u << 30);  // addr[56:32] | type=2
  i32x8 g1;
  g1[0] = (int)((2u << 16) | (1u << 20) |
                (pad_interval_code << 22) | (pad_amount_code << 25));
  g1[1] = (int)((tensor_d0 & 0xFFFFu) << 16);
  g1[2] = (int)((tensor_d0 >> 16) | ((tensor_d1 & 0xFFFFu) << 16));
  g1[3] = (int)((tensor_d1 >> 16) | (tile_d0 << 16));
  g1[4] = (int)tile_d1;                             // tile_dim1 | tile_dim2=0
  g1[5] = (int)row_stride;                          // tensor_dim0_stride[31:0]
  g1[6] = 0;
  g1[7] = 0;
  i32x4 gz = {0, 0, 0, 0};
#if defined(__clang_major__) && (__clang_major__ >= 23)
  i32x8 gz8 = {0, 0, 0, 0, 0, 0, 0, 0};
  __builtin_amdgcn_tensor_load_to_lds(g0, g1, gz, gz, gz8, 0);
#else
  __builtin_amdgcn_tensor_load_to_lds(g0, g1, gz, gz, 0);
#endif
}

__device__ __forceinline__ void tdm_issue_chunk(int buf, int kc, int rowBase,
                                                const float* feat, const float* W) {
  unsigned base = (unsigned)buf * (unsigned)(BUF_FLOATS * 4);
  // A tile: feat rows [rowBase, rowBase+128) x cols [kc*KC, kc*KC+KC)
  // pad: row = 16 DWORDs -> interval code 3 (2^4=16), amount 4 DWORDs -> code 3 => pitch 20
  tdm_load_2d(base, feat + (size_t)rowBase * IN_DIM + kc * KC,
              (unsigned)KC, (unsigned)(N_NODES - rowBase),
              (unsigned)KC, 128u, (unsigned)IN_DIM, 3u, 3u);
  // B tile: W rows [kc*KC, kc*KC+KC) x all 128 cols
  // pad: row = 128 DWORDs -> interval code 6 (2^7=128), amount 8 DWORDs -> code 7 => pitch 136
  tdm_load_2d(base + (unsigned)(128 * PA * 4), W + (size_t)(kc * KC) * HF,
              (unsigned)HF, (unsigned)(IN_DIM - kc * KC),
              (unsigned)HF, (unsigned)KC, (unsigned)HF, 6u, 7u);
}
#endif  // USE_TDM

// ---------------------------------------------------------------------
// Kernel 1: h[N,128] = feat[N,256] @ W[256,128] via V_WMMA_F32_16X16X4_F32
// block = 256 threads (8 waves). Block tile: 128 rows x 128 cols.
// Wave w computes rows [w*16, w*16+16) x all 128 cols (8 col tiles).
// TDM double-buffers the LDS staging (wave0 issues, s_wait_tensorcnt syncs).
// ---------------------------------------------------------------------
__global__ __launch_bounds__(256)
void gemm_node_proj(const float* __restrict__ feat, const float* __restrict__ W,
                    float* __restrict__ hbuf) {
  extern __shared__ float smem[];

  const int tid  = threadIdx.x;
  const int lane = tid & 31;
  const int wave = tid >> 5;
  const int rowBase = blockIdx.x * 128;
  const int m0   = wave * 16;
  const int mlo  = lane & 15;
  const int koff = (lane < 16) ? 0 : 2;   // 16x4 f32 A layout: half-wave selects K pair

  v8f acc[8];
  const v8f vzero = {0.f, 0.f, 0.f, 0.f, 0.f, 0.f, 0.f, 0.f};
#pragma unroll
  for (int t = 0; t < 8; ++t) acc[t] = vzero;

#if USE_TDM
  if (tid < 32) tdm_issue_chunk(0, 0, rowBase, feat, W);   // prologue: chunk 0 -> buf 0
#endif

  for (int kc = 0; kc < NCHUNK; ++kc) {
    float* As = smem + (kc & 1) * BUF_FLOATS;   // [128][PA]
    float* Bs = As + 128 * PA;                  // [KC][PB]

#if USE_TDM
    if (tid < 32) {
      if (kc + 1 < NCHUNK) {
        // issue next chunk into the buffer freed at the end of iteration kc-1
        tdm_issue_chunk((kc + 1) & 1, kc + 1, rowBase, feat, W);
        __builtin_amdgcn_s_wait_tensorcnt((short)2);   // chunk kc's 2 loads done (in-order)
      } else {
        __builtin_amdgcn_s_wait_tensorcnt((short)0);
      }
    }
    __syncthreads();
#else
    // ---- cooperative staging fallback (coalesced float4 loads) ----
#pragma unroll
    for (int i = 0; i < 2; ++i) {               // A: 128 x 16 floats = 512 float4
      int idx = tid + i * 256;
      int r   = idx >> 2;                       // 4 float4 per 16-float row
      int c   = (idx & 3) << 2;
      int gr  = rowBase + r;
      gr = gr < N_NODES ? gr : (N_NODES - 1);
      const float4 v = *(const float4*)(feat + (size_t)gr * IN_DIM + kc * KC + c);
      float* d = As + r * PA + c;
      d[0] = v.x; d[1] = v.y; d[2] = v.z; d[3] = v.w;
    }
#pragma unroll
    for (int i = 0; i < 2; ++i) {               // B: 16 x 128 floats = 512 float4
      int idx = tid + i * 256;
      int r   = idx >> 5;                       // 32 float4 per 128-float row
      int c   = (idx & 31) << 2;
      const float4 v = *(const float4*)(W + (size_t)(kc * KC + r) * HF + c);
      float* d = Bs + r * PB + c;
      d[0] = v.x; d[1] = v.y; d[2] = v.z; d[3] = v.w;
    }
    __syncthreads();
#endif

#if USE_WMMA
#pragma unroll
    for (int ks = 0; ks < KC / 4; ++ks) {
      const int kk = ks * 4 + koff;
      v2f afrag;
      const float* ap = As + (m0 + mlo) * PA + kk;      // A[m][kk], A[m][kk+1]
      afrag.x = ap[0]; afrag.y = ap[1];
#pragma unroll
      for (int ct = 0; ct < 8; ++ct) {
        v2f bfrag;
        const float* bp = Bs + kk * PB + ct * 16 + mlo; // B[kk][n], B[kk+1][n]
        bfrag.x = bp[0]; bfrag.y = bp[PB];
        acc[ct] = __builtin_amdgcn_wmma_f32_16x16x4_f32(
            false, afrag, false, bfrag, (short)0, acc[ct], false, false);
      }
    }
#else
    for (int k = 0; k < KC; ++k) {
#pragma unroll
      for (int ct = 0; ct < 8; ++ct) {
        float b = Bs[k * PB + ct * 16 + mlo];
#pragma unroll
        for (int v = 0; v < 8; ++v) {
          int m = v + ((lane < 16) ? 0 : 8);
          acc[ct][v] += As[(m0 + m) * PA + k] * b;
        }
      }
    }
#endif
    __syncthreads();   // all waves done reading this buffer before TDM reuses it
  }

  // C/D layout: vgpr v, lanes 0-15 -> M=v, lanes 16-31 -> M=v+8; N = lane%16
#pragma unroll
  for (int ct = 0; ct < 8; ++ct) {
#pragma unroll
    for (int v = 0; v < 8; ++v) {
      int m   = v + ((lane < 16) ? 0 : 8);
      int row = rowBase + m0 + m;
      if (row < N_NODES)
        hbuf[(size_t)row * HF + ct * 16 + mlo] = acc[ct][v];
    }
  }
}

// ------------------- Kernel 2: h_l / h_r per (node, head) -------------------
__global__ __launch_bounds__(256)
void node_attn(const float* __restrict__ hbuf, const float* __restrict__ attn_l,
               const float* __restrict__ attn_r, float* __restrict__ hl,
               float* __restrict__ hr) {
  int idx = blockIdx.x * blockDim.x + threadIdx.x;
  if (idx >= N_NODES * H_HEADS) return;
  int hh = idx & 3;
  const float* hp = hbuf + (size_t)(idx >> 2) * HF + hh * F_DIM;
  const float* al = attn_l + hh * F_DIM;
  const float* ar = attn_r + hh * F_DIM;
  float sl = 0.f, sr = 0.f;
#pragma unroll
  for (int f = 0; f < F_DIM; ++f) { float v = hp[f]; sl += v * al[f]; sr += v * ar[f]; }
  hl[idx] = sl; hr[idx] = sr;
}

// ---- Kernel 3: collapse edge branch to 8x4 table: h_e_type[t][h] ----
__global__ void edge_type_logits(const float* __restrict__ edge_emb,
                                 const float* __restrict__ W_e,
                                 const float* __restrict__ attn_e,
                                 float* __restrict__ h_e_type) {
  __shared__ float sh[NTYPES * H_HEADS];
  int c = threadIdx.x;                       // 0..255 = column of W_e
  if (c < NTYPES * H_HEADS) sh[c] = 0.f;
  __syncthreads();
  float ae = attn_e[c];
  int hh = c >> 6;                           // c / EF_DIM
  for (int t = 0; t < NTYPES; ++t) {
    float s = 0.f;
    for (int k = 0; k < EF_DIM; ++k)
      s += edge_emb[t * EF_DIM + k] * W_e[k * (H_HEADS * EF_DIM) + c];
    atomicAdd(&sh[t * H_HEADS + hh], s * ae);
  }
  __syncthreads();
  if (c < NTYPES * H_HEADS) h_e_type[c] = sh[c];
}

// ---- monotonic uint encoding of float for atomicMax over negatives ----
__device__ __forceinline__ unsigned encf(float x) {
  unsigned b = __float_as_uint(x);
  return (b & 0x80000000u) ? ~b : (b | 0x80000000u);
}
__device__ __forceinline__ float decf(unsigned u) {
  return __uint_as_float((u & 0x80000000u) ? (u & 0x7FFFFFFFu) : ~u);
}

__device__ __forceinline__ float edge_logit(int e, int hh, const int* __restrict__ src,
    const int* __restrict__ dst, const int* __restrict__ ef,
    const float* __restrict__ hl, const float* __restrict__ hr,
    const float* __restrict__ he, int& d) {
  int s = src[e]; d = dst[e]; int t = ef[e];
  float x = hl[s * 4 + hh] + hr[d * 4 + hh] + he[t * 4 + hh];
  return x > 0.f ? x : NEG_SLOPE * x;
}

// ------------------- Kernel 0: init accumulators -------------------
__global__ void init_buffers(float* __restrict__ out, unsigned* __restrict__ m_u,
                             float* __restrict__ s_sum) {
  int i = blockIdx.x * blockDim.x + threadIdx.x;
  if (i < N_NODES * HF) out[i] = 0.f;
  if (i < N_NODES * H_HEADS) { m_u[i] = 0u; s_sum[i] = 0.f; }
}

// ------------------- Kernel 4: segment max over dst -------------------
__global__ __launch_bounds__(256)
void edge_max(const int* __restrict__ src, const int* __restrict__ dst,
              const int* __restrict__ ef, const float* __restrict__ hl,
              const float* __restrict__ hr, const float* __restrict__ he,
              unsigned* __restrict__ m_u) {
  int idx = blockIdx.x * blockDim.x + threadIdx.x;
  if (idx >= N_EDGES * H_HEADS) return;
  int d; float x = edge_logit(idx >> 2, idx & 3, src, dst, ef, hl, hr, he, d);
  atomicMax(&m_u[d * 4 + (idx & 3)], encf(x));
}

// ------------------- Kernel 5: segment exp-sum -------------------
__global__ __launch_bounds__(256)
void edge_expsum(const int* __restrict__ src, const int* __restrict__ dst,
                 const int* __restrict__ ef, const float* __restrict__ hl,
                 const float* __restrict__ hr, const float* __restrict__ he,
                 const unsigned* __restrict__ m_u, float* __restrict__ s_sum) {
  int idx = blockIdx.x * blockDim.x + threadIdx.x;
  if (idx >= N_EDGES * H_HEADS) return;
  int hh = idx & 3, d;
  float x = edge_logit(idx >> 2, hh, src, dst, ef, hl, hr, he, d);
  float m = decf(m_u[d * 4 + hh]);
  atomicAdd(&s_sum[d * 4 + hh], __expf(x - m));
}

// --------- Kernel 6: a = softmax, write a, out[dst] += h[src]*a ---------
__global__ __launch_bounds__(256)
void edge_aggregate(const int* __restrict__ src, const int* __restrict__ dst,
                    const int* __restrict__ ef, const float* __restrict__ hl,
                    const float* __restrict__ hr, const float* __restrict__ he,
                    const unsigned* __restrict__ m_u, const float* __restrict__ s_sum,
                    const float* __restrict__ hbuf, float* __restrict__ out,
                    float* __restrict__ a_out) {
  int gid = blockIdx.x * blockDim.x + threadIdx.x;   // < E*128 = 204.8M
  if (gid >= N_EDGES * HF) return;
  int e = gid >> 7;
  int c = gid & 127;
  int hh = c >> 5, f = c & 31;
  int s = src[e], d = dst[e], t = ef[e];             // wave-uniform (32 lanes/edge-head)
  float x = hl[s * 4 + hh] + hr[d * 4 + hh] + he[t * 4 + hh];
  x = x > 0.f ? x : NEG_SLOPE * x;
  float m = decf(m_u[d * 4 + hh]);
  float a = __expf(x - m) / s_sum[d * 4 + hh];
  if (f == 0) a_out[e * 4 + hh] = a;
  atomicAdd(&out[(size_t)d * HF + c], hbuf[(size_t)s * HF + c] * a);
}

// ---------------------------------------------------------------------
extern "C" void kernel_launch(void* const* d_in, const int* in_sizes, int n_in,
                              void* d_out, int out_size, void* d_ws, size_t ws_size,
                              hipStream_t stream) {
  const float* feat     = (const float*)d_in[0];
  const int*   e_feat   = (const int*)  d_in[1];
  const int*   src      = (const int*)  d_in[2];
  const int*   dst      = (const int*)  d_in[3];
  const float* W        = (const float*)d_in[4];
  const float* W_e      = (const float*)d_in[5];
  const float* edge_emb = (const float*)d_in[6];
  const float* attn_l   = (const float*)d_in[7];
  const float* attn_r   = (const float*)d_in[8];
  const float* attn_e   = (const float*)d_in[9];

  float* out   = (float*)d_out;                      // [N,H,F]
  float* a_out = out + (size_t)N_NODES * HF;         // [E,H]

  // workspace layout (all 4-byte types): ~57.6 MB
  float*    hbuf  = (float*)d_ws;                          // N*HF
  float*    hl    = hbuf + (size_t)N_NODES * HF;           // N*H
  float*    hr    = hl + N_NODES * H_HEADS;                // N*H
  float*    he    = hr + N_NODES * H_HEADS;                // NTYPES*H
  unsigned* m_u   = (unsigned*)(he + NTYPES * H_HEADS);    // N*H
  float*    s_sum = (float*)(m_u + N_NODES * H_HEADS);     // N*H

  init_buffers<<<(N_NODES * HF + 255) / 256, 256, 0, stream>>>(out, m_u, s_sum);

  {
    dim3 grid((N_NODES + 127) / 128);
    size_t shmem = (size_t)(2 * BUF_FLOATS) * sizeof(float);   // 37888 B
    gemm_node_proj<<<grid, 256, shmem, stream>>>(feat, W, hbuf);
  }

  node_attn<<<(N_NODES * H_HEADS + 255) / 256, 256, 0, stream>>>(hbuf, attn_l, attn_r, hl, hr);
  edge_type_logits<<<1, 256, 0, stream>>>(edge_emb, W_e, attn_e, he);

  int eh = N_EDGES * H_HEADS;
  edge_max   <<<(eh + 255) / 256, 256, 0, stream>>>(src, dst, e_feat, hl, hr, he, m_u);
  edge_expsum<<<(eh + 255) / 256, 256, 0, stream>>>(src, dst, e_feat, hl, hr, he, m_u, s_sum);

  int ec = N_EDGES * HF;  // 204.8M < 2^31
  edge_aggregate<<<(unsigned)((ec + 255) / 256), 256, 0, stream>>>(
      src, dst, e_feat, hl, hr, he, m_u, s_sum, hbuf, out, a_out);
}